// ReasoningNet_46299747451451
// MI455X (gfx1250) — compile-verified
//
#include <hip/hip_runtime.h>
#include <hip/hip_bf16.h>

typedef __attribute__((ext_vector_type(2))) float v2f;
typedef __attribute__((ext_vector_type(8))) float v8f;

__device__ __forceinline__ v8f wmma4(v2f a, v2f b, v8f c) {
    // D = A(16x4 f32) * B(4x16 f32) + C(16x16 f32)
    return __builtin_amdgcn_wmma_f32_16x16x4_f32(
        /*neg_a=*/false, a, /*neg_b=*/false, b,
        /*c_mod=*/(short)0, c, /*reuse_a=*/false, /*reuse_b=*/false);
}

__global__ void zero_f32(float* __restrict__ p, int n) {
    int i = blockIdx.x * blockDim.x + threadIdx.x;
    if (i < n) p[i] = 0.0f;
}

// ---------------------------------------------------------------------------
// msg = ReLU(xprev[src] + edge_attr @ lin_w + lin_b); agg[dst] += msg
// One wave per 16-edge tile; lin_w staged in LDS in WMMA-B fragment layout so
// the inner loop is ds_load_b64 + v_wmma. Gathers are issued as a clause
// before each WMMA chain so their latency hides under the matrix ops.
// ---------------------------------------------------------------------------
__global__ void edge_conv(const float* __restrict__ xprev,  // [N,64]
                          const float* __restrict__ eattr,  // [E,32]
                          const int*   __restrict__ eidx,   // [2,E]
                          const float* __restrict__ lin_w,  // [32,64]
                          const float* __restrict__ lin_b,  // [64]
                          float* __restrict__ agg,          // [N,64]
                          int E) {
    __shared__ v2f sB[8 * 4 * 32];   // 8 KB

    for (int idx = threadIdx.x; idx < 8 * 4 * 32; idx += blockDim.x) {
        int ln   = idx & 31;
        int frag = idx >> 5;         // kk*4 + nt
        int kk   = frag >> 2;
        int nt   = frag & 3;
        int h    = ln >> 4;
        int c    = (nt << 4) + (ln & 15);
        int k    = (kk << 2) + (h << 1);
        v2f b;
        b.x = lin_w[k * 64 + c];
        b.y = lin_w[(k + 1) * 64 + c];
        sB[idx] = b;
    }
    __syncthreads();

    const int lane = threadIdx.x & 31;
    const int wave = threadIdx.x >> 5;
    const int l15  = lane & 15;
    const int half = lane >> 4;

    float bias[4];
#pragma unroll
    for (int nt = 0; nt < 4; ++nt) bias[nt] = lin_b[nt * 16 + l15];

    const int nTiles = (E + 15) >> 4;
    const int wavesPerBlock = blockDim.x >> 5;
    const int stride = gridDim.x * wavesPerBlock;
    for (int tile = blockIdx.x * wavesPerBlock + wave; tile < nTiles; tile += stride) {
        const int e0 = tile * 16;
        if (e0 + 16 <= E) {
            // ---- full tile: no per-lane guards anywhere ----
            const float* ap = eattr + (size_t)(e0 + l15) * 32 + half * 2;
            v2f Af[8];
#pragma unroll
            for (int kk = 0; kk < 8; ++kk) {
                Af[kk].x = ap[kk * 4];
                Af[kk].y = ap[kk * 4 + 1];
            }
            int sbase[8], dbase[8];
#pragma unroll
            for (int v = 0; v < 8; ++v) {
                int e = e0 + half * 8 + v;
                sbase[v] = eidx[e] * 64;
                dbase[v] = eidx[E + e] * 64;
            }
#pragma unroll
            for (int nt = 0; nt < 4; ++nt) {
                const int col = nt * 16 + l15;
                // issue the 8 gathers first: latency hides under the WMMA chain
                float xg[8];
#pragma unroll
                for (int v = 0; v < 8; ++v) xg[v] = xprev[sbase[v] + col];
                v8f acc;
#pragma unroll
                for (int i = 0; i < 8; ++i) acc[i] = bias[nt];
#pragma unroll
                for (int kk = 0; kk < 8; ++kk)
                    acc = wmma4(Af[kk], sB[(kk * 4 + nt) * 32 + lane], acc);
#pragma unroll
                for (int v = 0; v < 8; ++v) {
                    float m = acc[v] + xg[v];
                    m = m > 0.0f ? m : 0.0f;
                    atomicAdd(&agg[dbase[v] + col], m);
                }
            }
        } else {
            // ---- tail tile: clamp loads, predicate stores ----
            int arow = e0 + l15; if (arow >= E) arow = E - 1;
            const float* ap = eattr + (size_t)arow * 32 + half * 2;
            v2f Af[8];
#pragma unroll
            for (int kk = 0; kk < 8; ++kk) {
                Af[kk].x = ap[kk * 4];
                Af[kk].y = ap[kk * 4 + 1];
            }
            int sbase[8], dbase[8], evalid[8];
#pragma unroll
            for (int v = 0; v < 8; ++v) {
                int e = e0 + half * 8 + v;
                evalid[v] = (e < E);
                int ec = evalid[v] ? e : (E - 1);
                sbase[v] = eidx[ec] * 64;
                dbase[v] = eidx[E + ec] * 64;
            }
#pragma unroll
            for (int nt = 0; nt < 4; ++nt) {
                const int col = nt * 16 + l15;
                float xg[8];
#pragma unroll
                for (int v = 0; v < 8; ++v) xg[v] = xprev[sbase[v] + col];
                v8f acc;
#pragma unroll
                for (int i = 0; i < 8; ++i) acc[i] = bias[nt];
#pragma unroll
                for (int kk = 0; kk < 8; ++kk)
                    acc = wmma4(Af[kk], sB[(kk * 4 + nt) * 32 + lane], acc);
#pragma unroll
                for (int v = 0; v < 8; ++v) {
                    if (evalid[v]) {
                        float m = acc[v] + xg[v];
                        m = m > 0.0f ? m : 0.0f;
                        atomicAdd(&agg[dbase[v] + col], m);
                    }
                }
            }
        }
    }
}

// ---------------------------------------------------------------------------
// h = ReLU(ReLU((x + agg) @ w1 + b1) @ w2 + b2)   (all 64x64)
// Weights staged in LDS in fragment layout: ((l*16 + kk)*4 + nt)*32 + lane.
// ---------------------------------------------------------------------------
__global__ void node_mlp(const float* __restrict__ x,    // [N,64]
                         const float* __restrict__ agg,  // [N,64]
                         const float* __restrict__ w1, const float* __restrict__ b1,
                         const float* __restrict__ w2, const float* __restrict__ b2,
                         float* __restrict__ hout,       // [N,64]
                         int N) {
    __shared__ v2f   sW[2 * 16 * 4 * 32];   // 32 KB, fragment layout
    __shared__ float sH[4][16 * 65];        // per-wave transpose staging

    for (int idx = threadIdx.x; idx < 2 * 16 * 4 * 32; idx += blockDim.x) {
        int ln   = idx & 31;
        int frag = idx >> 5;
        int nt   = frag & 3;
        int kk   = (frag >> 2) & 15;
        int l    = frag >> 6;
        int h    = ln >> 4;
        int c    = (nt << 4) + (ln & 15);
        int k    = (kk << 2) + (h << 1);
        const float* w = l ? w2 : w1;
        v2f b;
        b.x = w[k * 64 + c];
        b.y = w[(k + 1) * 64 + c];
        sW[idx] = b;
    }
    __syncthreads();

    const int lane = threadIdx.x & 31;
    const int wave = threadIdx.x >> 5;
    const int l15  = lane & 15;
    const int half = lane >> 4;
    float* myH = sH[wave];

    float bias1[4], bias2[4];
#pragma unroll
    for (int nt = 0; nt < 4; ++nt) {
        bias1[nt] = b1[nt * 16 + l15];
        bias2[nt] = b2[nt * 16 + l15];
    }

    const int nTiles = (N + 15) >> 4;
    const int wavesPerBlock = blockDim.x >> 5;
    const int stride = gridDim.x * wavesPerBlock;
    for (int tile = blockIdx.x * wavesPerBlock + wave; tile < nTiles; tile += stride) {
        const int n0 = tile * 16;
        int row = n0 + l15; if (row >= N) row = N - 1;
        const float* xr = x + (size_t)row * 64;
        const float* ar = agg + (size_t)row * 64;
        v2f Af[16];
#pragma unroll
        for (int kk = 0; kk < 16; ++kk) {
            int k = kk * 4 + half * 2;
            Af[kk].x = xr[k] + ar[k];
            Af[kk].y = xr[k + 1] + ar[k + 1];
        }
        // layer 1 -> ReLU -> LDS (D layout -> row-major tile)
#pragma unroll
        for (int nt = 0; nt < 4; ++nt) {
            const int col = nt * 16 + l15;
            v8f acc;
#pragma unroll
            for (int i = 0; i < 8; ++i) acc[i] = bias1[nt];
#pragma unroll
            for (int kk = 0; kk < 16; ++kk)
                acc = wmma4(Af[kk], sW[(kk * 4 + nt) * 32 + lane], acc);
#pragma unroll
            for (int v = 0; v < 8; ++v) {
                float r = acc[v];
                myH[(half * 8 + v) * 65 + col] = r > 0.0f ? r : 0.0f;
            }
        }
        // layer 2: A from LDS tile (DS ops within a wave are in-order)
        v2f A2[16];
#pragma unroll
        for (int kk = 0; kk < 16; ++kk) {
            int k = kk * 4 + half * 2;
            A2[kk].x = myH[l15 * 65 + k];
            A2[kk].y = myH[l15 * 65 + k + 1];
        }
#pragma unroll
        for (int nt = 0; nt < 4; ++nt) {
            const int col = nt * 16 + l15;
            v8f acc;
#pragma unroll
            for (int i = 0; i < 8; ++i) acc[i] = bias2[nt];
#pragma unroll
            for (int kk = 0; kk < 16; ++kk)
                acc = wmma4(A2[kk], sW[(16 * 4 + kk * 4 + nt) * 32 + lane], acc);
            if (n0 + 16 <= N) {
#pragma unroll
                for (int v = 0; v < 8; ++v) {
                    float r = acc[v];
                    hout[(size_t)(n0 + half * 8 + v) * 64 + col] = r > 0.0f ? r : 0.0f;
                }
            } else {
#pragma unroll
                for (int v = 0; v < 8; ++v) {
                    int orow = n0 + half * 8 + v;
                    if (orow < N) {
                        float r = acc[v];
                        hout[(size_t)orow * 64 + col] = r > 0.0f ? r : 0.0f;
                    }
                }
            }
        }
    }
}

// ---------------------------------------------------------------------------
// out[N,32] = h[N,64] @ w[64,32] + b   (head_w staged in LDS fragment layout)
// ---------------------------------------------------------------------------
__global__ void head_kernel(const float* __restrict__ h, const float* __restrict__ w,
                            const float* __restrict__ b, float* __restrict__ out, int N) {
    __shared__ v2f sB[16 * 2 * 32];   // 8 KB

    for (int idx = threadIdx.x; idx < 16 * 2 * 32; idx += blockDim.x) {
        int ln   = idx & 31;
        int frag = idx >> 5;
        int nt   = frag & 1;
        int kk   = frag >> 1;
        int hf   = ln >> 4;
        int c    = (nt << 4) + (ln & 15);
        int k    = (kk << 2) + (hf << 1);
        v2f bv;
        bv.x = w[k * 32 + c];
        bv.y = w[(k + 1) * 32 + c];
        sB[idx] = bv;
    }
    __syncthreads();

    const int lane = threadIdx.x & 31;
    const int wave = threadIdx.x >> 5;
    const int l15  = lane & 15;
    const int half = lane >> 4;

    float bias[2] = { b[l15], b[16 + l15] };

    const int nTiles = (N + 15) >> 4;
    const int wavesPerBlock = blockDim.x >> 5;
    const int stride = gridDim.x * wavesPerBlock;
    for (int tile = blockIdx.x * wavesPerBlock + wave; tile < nTiles; tile += stride) {
        const int n0 = tile * 16;
        int row = n0 + l15; if (row >= N) row = N - 1;
        const float* hr = h + (size_t)row * 64;
        v2f Af[16];
#pragma unroll
        for (int kk = 0; kk < 16; ++kk) {
            int k = kk * 4 + half * 2;
            Af[kk].x = hr[k];
            Af[kk].y = hr[k + 1];
        }
#pragma unroll
        for (int nt = 0; nt < 2; ++nt) {
            v8f acc;
#pragma unroll
            for (int i = 0; i < 8; ++i) acc[i] = bias[nt];
#pragma unroll
            for (int kk = 0; kk < 16; ++kk)
                acc = wmma4(Af[kk], sB[(kk * 2 + nt) * 32 + lane], acc);
            const int col = nt * 16 + l15;
            if (n0 + 16 <= N) {
#pragma unroll
                for (int v = 0; v < 8; ++v)
                    out[(size_t)(n0 + half * 8 + v) * 32 + col] = acc[v];
            } else {
#pragma unroll
                for (int v = 0; v < 8; ++v) {
                    int orow = n0 + half * 8 + v;
                    if (orow < N) out[(size_t)orow * 32 + col] = acc[v];
                }
            }
        }
    }
}

extern "C" void kernel_launch(void* const* d_in, const int* in_sizes, int n_in,
                              void* d_out, int out_size, void* d_ws, size_t ws_size,
                              hipStream_t stream) {
    const float* x        = (const float*)d_in[0];
    const float* eattr    = (const float*)d_in[1];
    const int*   eidx     = (const int*)  d_in[2];
    const float* c1_lin_w = (const float*)d_in[3];
    const float* c1_lin_b = (const float*)d_in[4];
    const float* c1_w1    = (const float*)d_in[5];
    const float* c1_b1    = (const float*)d_in[6];
    const float* c1_w2    = (const float*)d_in[7];
    const float* c1_b2    = (const float*)d_in[8];
    const float* c2_lin_w = (const float*)d_in[9];
    const float* c2_lin_b = (const float*)d_in[10];
    const float* c2_w1    = (const float*)d_in[11];
    const float* c2_b1    = (const float*)d_in[12];
    const float* c2_w2    = (const float*)d_in[13];
    const float* c2_b2    = (const float*)d_in[14];
    const float* head_w   = (const float*)d_in[15];
    const float* head_b   = (const float*)d_in[16];

    const int N = in_sizes[0] / 64;
    const int E = in_sizes[1] / 32;

    float* ws  = (float*)d_ws;
    float* agg = ws;                       // [N,64]
    float* h1  = ws + (size_t)N * 64;      // [N,64]
    float* h2  = ws + (size_t)2 * N * 64;  // [N,64]

    const int nodeElems = N * 64;
    const int zBlocks = (nodeElems + 255) / 256;

    const int edgeTiles = (E + 15) / 16;
    int edgeBlocks = (edgeTiles + 7) / 8;       // 8 waves/block @ 256 threads
    if (edgeBlocks > 1024) edgeBlocks = 1024;   // grid-stride amortizes staging

    const int nodeTiles = (N + 15) / 16;
    int mlpBlocks = (nodeTiles + 3) / 4;        // 4 waves/block @ 128 threads
    if (mlpBlocks > 800) mlpBlocks = 800;

    // conv1
    zero_f32<<<zBlocks, 256, 0, stream>>>(agg, nodeElems);
    edge_conv<<<edgeBlocks, 256, 0, stream>>>(x, eattr, eidx, c1_lin_w, c1_lin_b, agg, E);
    node_mlp<<<mlpBlocks, 128, 0, stream>>>(x, agg, c1_w1, c1_b1, c1_w2, c1_b2, h1, N);
    // conv2
    zero_f32<<<zBlocks, 256, 0, stream>>>(agg, nodeElems);
    edge_conv<<<edgeBlocks, 256, 0, stream>>>(h1, eattr, eidx, c2_lin_w, c2_lin_b, agg, E);
    node_mlp<<<mlpBlocks, 128, 0, stream>>>(h1, agg, c2_w1, c2_b1, c2_w2, c2_b2, h2, N);
    // head
    head_kernel<<<mlpBlocks, 128, 0, stream>>>(h2, head_w, head_b, (float*)d_out, N);
}